// TokenChoiceTopKRouter_70136815943760
// MI455X (gfx1250) — compile-verified
//
#include <hip/hip_runtime.h>
#include <hip/hip_bf16.h>

typedef __attribute__((ext_vector_type(2))) float v2f;
typedef __attribute__((ext_vector_type(8))) float v8f;

#define N_TOKENS   32768
#define DIM        4096
#define N_EXPERTS  64
#define ROUTE_SCALE 1.0f
#define N_FLAT     (N_TOKENS * 2)      // 65536 flat (token, slot) entries
#define N_BLOCKS   512                 // kernel A/C blocks; 128 flat entries each

// ---------------------------------------------------------------------------
// Kernel A: fp32 WMMA GEMM (16 tokens x 64 experts per wave) + softmax + top2
// + per-workgroup expert histogram.
// Block = 128 threads (4 waves). Wave w handles tokens
// [blockIdx*64 + w*16, +16). Flat entries covered by a block: [blockIdx*128,+128).
// ---------------------------------------------------------------------------
__global__ void __launch_bounds__(128)
router_gemm_top2_kernel(const float* __restrict__ x,
                        const float* __restrict__ W,
                        float* __restrict__ ws_scores,
                        int*   __restrict__ ws_experts,
                        int*   __restrict__ blockHist)
{
    __shared__ float s_logits[4 * 16 * 64];   // 16KB: per-wave 16 tokens x 64 experts
    __shared__ int   s_hist[N_EXPERTS];

    const int tid  = threadIdx.x;
    const int wave = tid >> 5;
    const int lane = tid & 31;
    const int half = lane >> 4;      // selects K pair (ISA 32-bit A/B layout)
    const int lm   = lane & 15;      // M for A, N for B
    const int rowBase = blockIdx.x * 64 + wave * 16;

    if (tid < N_EXPERTS) s_hist[tid] = 0;

    // A: lane holds x[rowBase+lm][k + 2*half + {0,1}]  (v_wmma A 16x4 layout)
    // B: lane holds W[lm + 16t][k + 2*half + {0,1}]    (B 4x16 layout, B=W^T)
    const float* xrow = x + (size_t)(rowBase + lm) * DIM + 2 * half;
    const float* wrow = W + (size_t)lm * DIM + 2 * half;

    v8f acc0 = {}, acc1 = {}, acc2 = {}, acc3 = {};
    #pragma unroll 4
    for (int k = 0; k < DIM; k += 4) {
        v2f a  = *(const v2f*)(xrow + k);
        v2f b0 = *(const v2f*)(wrow + k);
        v2f b1 = *(const v2f*)(wrow + 16 * (size_t)DIM + k);
        v2f b2 = *(const v2f*)(wrow + 32 * (size_t)DIM + k);
        v2f b3 = *(const v2f*)(wrow + 48 * (size_t)DIM + k);
        acc0 = __builtin_amdgcn_wmma_f32_16x16x4_f32(false, a, false, b0, (short)0, acc0, false, false);
        acc1 = __builtin_amdgcn_wmma_f32_16x16x4_f32(false, a, false, b1, (short)0, acc1, false, false);
        acc2 = __builtin_amdgcn_wmma_f32_16x16x4_f32(false, a, false, b2, (short)0, acc2, false, false);
        acc3 = __builtin_amdgcn_wmma_f32_16x16x4_f32(false, a, false, b3, (short)0, acc3, false, false);
    }

    // C/D layout: VGPR r, lane half -> token = r + 8*half; N = lm; tile t -> expert lm+16t
    float* sl = s_logits + wave * (16 * 64);
    #pragma unroll
    for (int r = 0; r < 8; ++r) {
        const int tok = r + 8 * half;
        sl[tok * 64 + lm +  0] = acc0[r];
        sl[tok * 64 + lm + 16] = acc1[r];
        sl[tok * 64 + lm + 32] = acc2[r];
        sl[tok * 64 + lm + 48] = acc3[r];
    }
    __syncthreads();

    if (lane < 16) {
        const int token = rowBase + lane;
        const float* lp = s_logits + wave * (16 * 64) + lane * 64;

        float m = lp[0];
        #pragma unroll 8
        for (int e = 1; e < N_EXPERTS; ++e) m = fmaxf(m, lp[e]);

        float sum = 0.0f;
        float v1 = -__builtin_inff(), v2 = -__builtin_inff();
        int   i1 = 0, i2 = 0;
        for (int e = 0; e < N_EXPERTS; ++e) {
            const float l = lp[e];
            sum += __expf(l - m);
            if (l > v1)      { v2 = v1; i2 = i1; v1 = l; i1 = e; }
            else if (l > v2) { v2 = l; i2 = e; }
        }
        const float inv = 1.0f / sum;
        const float s1 = __expf(v1 - m) * inv;
        const float s2 = __expf(v2 - m) * inv;

        const int flat = 2 * token;
        ws_scores[flat]     = s1;  ws_experts[flat]     = i1;
        ws_scores[flat + 1] = s2;  ws_experts[flat + 1] = i2;
        atomicAdd(&s_hist[i1], 1);
        atomicAdd(&s_hist[i2], 1);
    }
    __syncthreads();
    if (tid < N_EXPERTS) blockHist[blockIdx.x * N_EXPERTS + tid] = s_hist[tid];
}

// ---------------------------------------------------------------------------
// Kernel B: expert totals (-> counts output), exclusive prefix over experts,
// and per-expert running offsets for each of the 512 blocks.
// ---------------------------------------------------------------------------
__global__ void __launch_bounds__(64)
router_scan_kernel(const int* __restrict__ blockHist,
                   int*       __restrict__ blockOff,
                   float*     __restrict__ counts_out)
{
    __shared__ int s_tot[N_EXPERTS];
    __shared__ int s_start[N_EXPERTS];

    const int e = threadIdx.x;
    int tot = 0;
    for (int b = 0; b < N_BLOCKS; ++b) tot += blockHist[b * N_EXPERTS + e];
    s_tot[e] = tot;
    counts_out[e] = (float)tot;
    __syncthreads();

    if (e == 0) {
        int run = 0;
        for (int i = 0; i < N_EXPERTS; ++i) { s_start[i] = run; run += s_tot[i]; }
    }
    __syncthreads();

    int off = s_start[e];
    for (int b = 0; b < N_BLOCKS; ++b) {
        blockOff[b * N_EXPERTS + e] = off;
        off += blockHist[b * N_EXPERTS + e];
    }
}

// ---------------------------------------------------------------------------
// Kernel C: stable scatter. Intra-block rank = #earlier entries (thread order
// == flat order) with same expert; global position = blockOff + rank.
// ---------------------------------------------------------------------------
__global__ void __launch_bounds__(128)
router_scatter_kernel(const float* __restrict__ ws_scores,
                      const int*   __restrict__ ws_experts,
                      const int*   __restrict__ blockOff,
                      float*       __restrict__ out_scores,
                      float*       __restrict__ out_indices)
{
    __shared__ int s_e[128];
    const int i    = threadIdx.x;
    const int flat = blockIdx.x * 128 + i;
    const int e    = ws_experts[flat];
    s_e[i] = e;
    __syncthreads();

    int rank = 0;
    for (int j = 0; j < i; ++j) rank += (s_e[j] == e) ? 1 : 0;

    const int pos = blockOff[blockIdx.x * N_EXPERTS + e] + rank;
    out_scores[pos]  = ws_scores[flat] * ROUTE_SCALE;
    out_indices[pos] = (float)(flat >> 1);   // perm // TOP_K == token index
}

// ---------------------------------------------------------------------------
extern "C" void kernel_launch(void* const* d_in, const int* in_sizes, int n_in,
                              void* d_out, int out_size, void* d_ws, size_t ws_size,
                              hipStream_t stream) {
    const float* x = (const float*)d_in[0];   // [32768, 4096] fp32
    const float* W = (const float*)d_in[1];   // [64, 4096] fp32
    float* out = (float*)d_out;               // scores | token indices | counts

    // Workspace layout (deterministic, fully rewritten every call):
    char* ws = (char*)d_ws;
    float* ws_scores  = (float*)ws;                                   // 65536 f32
    int*   ws_experts = (int*)(ws + (size_t)N_FLAT * 4);              // 65536 i32
    int*   blockHist  = (int*)(ws + (size_t)N_FLAT * 8);              // 512*64 i32
    int*   blockOff   = (int*)(ws + (size_t)N_FLAT * 8
                                  + (size_t)N_BLOCKS * N_EXPERTS * 4);// 512*64 i32

    router_gemm_top2_kernel<<<N_BLOCKS, 128, 0, stream>>>(x, W, ws_scores, ws_experts, blockHist);
    router_scan_kernel<<<1, 64, 0, stream>>>(blockHist, blockOff, out + 2 * N_FLAT);
    router_scatter_kernel<<<N_BLOCKS, 128, 0, stream>>>(ws_scores, ws_experts, blockOff,
                                                        out, out + N_FLAT);
}